// Involution_463856468143
// MI455X (gfx1250) — compile-verified
//
#include <hip/hip_runtime.h>

// CDNA5 / gfx1250 fused involution kernel.
// HBM-bound (~420MB traffic, dominated by the per-pixel kernel output);
// GEMM part (M=2.1M, K=16, N=49) done with V_WMMA_F32_16X16X4_F32:
// 4 K-steps x 4 N-tiles per 16-pixel group.
// Outputs stored non-temporal (TH=NT) so the streaming 411MB kern write
// doesn't evict the reusable x rows (re-read by 7 vertically adjacent
// row-blocks via the 7x7 halo) from the 192MB L2.

typedef __attribute__((ext_vector_type(2))) float v2f;
typedef __attribute__((ext_vector_type(8))) float v8f;

#define K_SZ   7
#define KK     49
#define HID    16
#define H_IMG  256
#define W_IMG  256
#define B_IMG  32
#define BN_EPS 1e-5f

#define PIX_PER_WAVE  16
#define PIX_PER_BLOCK 128         // 8 waves * 16 pixels
#define TILE_W        (PIX_PER_BLOCK + 6)   // 134 (3-halo each side)
#define TILE_W_PAD    136
#define TILE_H        7

__global__ __launch_bounds__(256)
void involution_wmma_kernel(const float* __restrict__ x,
                            const float* __restrict__ w1,
                            const float* __restrict__ b1,
                            const float* __restrict__ bn_g,
                            const float* __restrict__ bn_b,
                            const float* __restrict__ bn_m,
                            const float* __restrict__ bn_v,
                            const float* __restrict__ w2,
                            const float* __restrict__ b2,
                            float* __restrict__ out,
                            float* __restrict__ kern_out)
{
    __shared__ float xtile[TILE_H * TILE_W_PAD];

    const int tid  = threadIdx.x;
    const int lane = tid & 31;
    const int wave = tid >> 5;
    const int half = lane >> 4;   // 0: lanes 0-15, 1: lanes 16-31
    const int l16  = lane & 15;

    const int blk   = blockIdx.x;
    const int wseg  = blk & 1;                // which 128-pixel segment of the row
    const int row   = (blk >> 1) & (H_IMG - 1);
    const int b     = blk >> 9;               // / (H * 2)
    const int wbase = wseg * PIX_PER_BLOCK;

    // ---- cooperative load of zero-padded x halo tile (7 rows x 134 cols) ----
    const float* xb = x + (size_t)b * (H_IMG * W_IMG);
    for (int i = tid; i < TILE_H * TILE_W; i += 256) {
        int yr = i / TILE_W;
        int xc = i - yr * TILE_W;
        int gy = row - 3 + yr;
        int gx = wbase - 3 + xc;
        float v = 0.0f;
        if (gy >= 0 && gy < H_IMG && gx >= 0 && gx < W_IMG)
            v = xb[gy * W_IMG + gx];
        xtile[yr * TILE_W_PAD + xc] = v;
    }
    __syncthreads();

    // ---- fold BN into per-channel affine: h_c = relu(alpha_c*x + beta_c) ----
    // This lane's channels per K-step s: c = 4s + 2*half + j, j in {0,1}
    float alpha[8], beta[8];
#pragma unroll
    for (int s = 0; s < 4; ++s) {
#pragma unroll
        for (int j = 0; j < 2; ++j) {
            int c = 4 * s + 2 * half + j;
            float sc = bn_g[c] * rsqrtf(bn_v[c] + BN_EPS);
            alpha[2 * s + j] = w1[c] * sc;
            beta[2 * s + j]  = (b1[c] - bn_m[c]) * sc + bn_b[c];
        }
    }

    // ---- B matrix (W2^T) registers: B[K=c, N=k] = w2[k, c]; zero-pad k>=49 ----
    v2f Bm[4][4];
#pragma unroll
    for (int t = 0; t < 4; ++t) {
        int k = 16 * t + l16;
        bool kv = (k < KK);
#pragma unroll
        for (int s = 0; s < 4; ++s) {
            int c0 = 4 * s + 2 * half;
            Bm[t][s].x = kv ? w2[k * HID + c0]     : 0.0f;
            Bm[t][s].y = kv ? w2[k * HID + c0 + 1] : 0.0f;
        }
    }

    // ---- accumulators initialized with b2 bias (N = l16 column per lane) ----
    v8f acc[4];
#pragma unroll
    for (int t = 0; t < 4; ++t) {
        int k = 16 * t + l16;
        float bv = (k < KK) ? b2[k] : 0.0f;
#pragma unroll
        for (int r = 0; r < 8; ++r) acc[t][r] = bv;
    }

    // ---- build A (16 pixels x 16 channels) and run 4x4 WMMAs ----
    const int wloc = wave * PIX_PER_WAVE;          // wave's pixel base in block
    const float xv = xtile[3 * TILE_W_PAD + 3 + wloc + l16];  // center x of A-row M=l16
#pragma unroll
    for (int s = 0; s < 4; ++s) {
        v2f a;
        a.x = fmaxf(fmaf(alpha[2 * s],     xv, beta[2 * s]),     0.0f);
        a.y = fmaxf(fmaf(alpha[2 * s + 1], xv, beta[2 * s + 1]), 0.0f);
#pragma unroll
        for (int t = 0; t < 4; ++t) {
            acc[t] = __builtin_amdgcn_wmma_f32_16x16x4_f32(
                /*neg_a=*/false, a, /*neg_b=*/false, Bm[t][s],
                /*c_mod=*/(short)0, acc[t],
                /*reuse_a=*/false, /*reuse_b=*/false);
        }
    }

    // ---- store per-pixel kernels (coalesced: k contiguous, 16-wide runs) ----
    // Non-temporal: 411MB streaming output, never re-read this dispatch.
    const size_t pix_base = ((size_t)b * H_IMG + row) * W_IMG + wbase + wloc;
#pragma unroll
    for (int t = 0; t < 4; ++t) {
        int k = 16 * t + l16;
        if (k < KK) {
#pragma unroll
            for (int r = 0; r < 8; ++r) {
                size_t pg = pix_base + r + 8 * half;     // D row -> pixel
                __builtin_nontemporal_store(acc[t][r], &kern_out[pg * KK + k]);
            }
        }
    }

    // ---- out[p] = sum_k kern[p,k] * patch(p,k), patch from LDS halo ----
#pragma unroll
    for (int r = 0; r < 8; ++r) {
        int pix = r + 8 * half;          // pixel index within wave (0..15)
        float p = 0.0f;
#pragma unroll
        for (int t = 0; t < 4; ++t) {
            int k = 16 * t + l16;
            float pv = 0.0f;
            if (k < KK) {
                int dy = k / K_SZ - 3;
                int dx = k % K_SZ - 3;
                pv = xtile[(3 + dy) * TILE_W_PAD + 3 + wloc + pix + dx];
            }
            p = fmaf(acc[t][r], pv, p);
        }
        // reduce over the 16 lanes of this half (k dimension)
        p += __shfl_xor(p, 1, 32);
        p += __shfl_xor(p, 2, 32);
        p += __shfl_xor(p, 4, 32);
        p += __shfl_xor(p, 8, 32);
        if (l16 == pix)                  // one lane per pixel stores
            __builtin_nontemporal_store(p, &out[pix_base + pix]);
    }
}

extern "C" void kernel_launch(void* const* d_in, const int* in_sizes, int n_in,
                              void* d_out, int out_size, void* d_ws, size_t ws_size,
                              hipStream_t stream) {
    (void)in_sizes; (void)n_in; (void)d_ws; (void)ws_size; (void)out_size;
    const float* x    = (const float*)d_in[0];
    const float* w1   = (const float*)d_in[1];
    const float* b1   = (const float*)d_in[2];
    const float* bn_g = (const float*)d_in[3];
    const float* bn_b = (const float*)d_in[4];
    const float* bn_m = (const float*)d_in[5];
    const float* bn_v = (const float*)d_in[6];
    const float* w2   = (const float*)d_in[7];
    const float* b2   = (const float*)d_in[8];

    float* out  = (float*)d_out;
    float* kern = out + (size_t)B_IMG * H_IMG * W_IMG;   // tuple element 1

    dim3 grid(B_IMG * H_IMG * (W_IMG / PIX_PER_BLOCK));  // 32*256*2 = 16384
    dim3 block(256);
    involution_wmma_kernel<<<grid, block, 0, stream>>>(
        x, w1, b1, bn_g, bn_b, bn_m, bn_v, w2, b2, out, kern);
}